// H_ATT_38723425140810
// MI455X (gfx1250) — compile-verified
//
#include <hip/hip_runtime.h>
#include <hip/hip_bf16.h>
#include <math.h>
#include <stdint.h>

// Problem constants (match reference)
#define B_  128
#define R_  64
#define D_  512
#define IN_ 1024

typedef __attribute__((ext_vector_type(2))) float v2f;
typedef __attribute__((ext_vector_type(8))) float v8f;

// D = A(16x4 f32) * B(4x16 f32) + C(16x16 f32)  -> v_wmma_f32_16x16x4_f32
__device__ __forceinline__ v8f wmma4(v2f a, v2f b, v8f c) {
  return __builtin_amdgcn_wmma_f32_16x16x4_f32(
      false, a, false, b, (short)0, c, false, false);
}

// CDNA5 async copy: 16 bytes global -> LDS per lane, tracked by ASYNCcnt.
__device__ __forceinline__ void async_copy16(uint32_t lds_addr, const float* gp) {
  asm volatile("global_load_async_to_lds_b128 %0, %1, off"
               :: "v"(lds_addr), "v"((unsigned long long)(uintptr_t)gp)
               : "memory");
}
__device__ __forceinline__ void wait_async0() {
  asm volatile("s_wait_asynccnt 0x0" ::: "memory");
}
__device__ __forceinline__ uint32_t lds_off(const void* p) {
  return (uint32_t)(uintptr_t)p;   // low 32 bits of generic LDS ptr = LDS offset
}

// ---------------------------------------------------------------------------
// Kernel 1: gated transform  Out = tanh(X@Wy^T + by) * sigmoid(X@Wg^T + bg)
// X: [8192, 1024], Wy/Wg: [512, 1024].
// Block = 4 waves covers 64(m) x 32(n). K staged in 32-float chunks via
// GLOBAL_LOAD_ASYNC_TO_LDS_B128 into DOUBLE-BUFFERED LDS: copies for chunk
// c+1 are issued before computing chunk c, so HBM/L2 latency overlaps the
// 32 WMMAs per chunk. Each wave holds 2 m-subtiles x {y,g} = 4 v8f accums.
// ---------------------------------------------------------------------------
#define KC   32
#define KCP  34   // padded LDS row stride (floats), even => 8B-aligned v2f

__global__ void gated_gemm_kernel(const float* __restrict__ X,
                                  const float* __restrict__ Wy,
                                  const float* __restrict__ by,
                                  const float* __restrict__ Wg,
                                  const float* __restrict__ bg,
                                  float* __restrict__ Out) {
  __shared__ float XS[2][64 * KCP];
  __shared__ float WyS[2][32 * KCP];
  __shared__ float WgS[2][32 * KCP];

  const int tid  = threadIdx.x;
  const int lane = tid & 31;
  const int wave = tid >> 5;
  const int mw   = wave >> 1;          // 0..1 : which 32-row half
  const int nw   = wave & 1;           // 0..1 : which 16-col subtile

  const int m0 = blockIdx.y * 64;      // 128 m-blocks
  const int n0 = blockIdx.x * 32;      // 16  n-blocks

  const int half = lane >> 4;          // ISA A/B K-half select
  const int l    = lane & 15;
  const int koff = half * 2;

  // Issue one chunk's worth of async copies into buffer `bsel`.
  auto stage = [&](int bsel, int k0) {
    float* xs = &XS[bsel][0];
    float* ws = &WyS[bsel][0];
    float* gs = &WgS[bsel][0];
#pragma unroll
    for (int i = 0; i < 4; ++i) {      // X: 64 rows x 8 16B-pieces = 512
      const int p = i * 128 + tid;
      const int r = p >> 3, c4 = (p & 7) * 4;
      async_copy16(lds_off(xs + r * KCP + c4),
                   X + (size_t)(m0 + r) * IN_ + k0 + c4);
    }
#pragma unroll
    for (int i = 0; i < 2; ++i) {      // Wy/Wg: 32 rows x 8 pieces = 256 each
      const int p = i * 128 + tid;
      const int r = p >> 3, c4 = (p & 7) * 4;
      async_copy16(lds_off(ws + r * KCP + c4),
                   Wy + (size_t)(n0 + r) * IN_ + k0 + c4);
      async_copy16(lds_off(gs + r * KCP + c4),
                   Wg + (size_t)(n0 + r) * IN_ + k0 + c4);
    }
  };

  v8f ay0 = {}, ag0 = {}, ay1 = {}, ag1 = {};

  const int nchunks = IN_ / KC;        // 32
  stage(0, 0);                         // prologue: fill buffer 0

  for (int c = 0; c < nchunks; ++c) {
    const int buf = c & 1;
    wait_async0();                     // this wave's copies for chunk c done
    __syncthreads();                   // all waves' copies visible
    if (c + 1 < nchunks)               // overlap: prefetch next chunk
      stage(buf ^ 1, (c + 1) * KC);

    const float* xa0 = &XS[buf][(mw * 32 + 0  + l) * KCP + koff];
    const float* xa1 = &XS[buf][(mw * 32 + 16 + l) * KCP + koff];
    const float* wyp = &WyS[buf][(nw * 16 + l) * KCP + koff];
    const float* wgp = &WgS[buf][(nw * 16 + l) * KCP + koff];
#pragma unroll
    for (int kk = 0; kk < KC; kk += 4) {
      v2f a0 = *(const v2f*)(xa0 + kk);
      v2f a1 = *(const v2f*)(xa1 + kk);
      v2f bY = *(const v2f*)(wyp + kk);
      v2f bG = *(const v2f*)(wgp + kk);
      ay0 = wmma4(a0, bY, ay0);
      ag0 = wmma4(a0, bG, ag0);
      ay1 = wmma4(a1, bY, ay1);
      ag1 = wmma4(a1, bG, ag1);
    }
    __syncthreads();                   // reads of `buf` done before it is
  }                                    // overwritten by the next iteration's prefetch

  const int nc   = n0 + nw * 16 + l;
  const float biy = by[nc];
  const float big = bg[nc];
  // C/D layout: VGPR j -> m = base + j + 8*half, n = nc
#pragma unroll
  for (int j = 0; j < 8; ++j) {
    const int r0 = m0 + mw * 32 + j + half * 8;
    float y0 = ay0[j] + biy, g0 = ag0[j] + big;
    float y1 = ay1[j] + biy, g1 = ag1[j] + big;
    Out[(size_t)(r0)      * D_ + nc] = tanhf(y0) * (1.0f / (1.0f + expf(-g0)));
    Out[(size_t)(r0 + 16) * D_ + nc] = tanhf(y1) * (1.0f / (1.0f + expf(-g1)));
  }
}

// ---------------------------------------------------------------------------
// Kernel 2: scores[b,r,c] = (sum_d q*h*w) / max(||q*h||,eps) + ba
// Two fused batched GEMMs: (q*w)@h^T and (q*q)@(h*h)^T over D=512.
// One block (4 waves) per b; each wave computes a 32x32 quadrant
// (2x2 subtiles x 2 accumulators = 8 WMMAs per K-step on 5 b64 loads).
// ---------------------------------------------------------------------------
__global__ void score_kernel(const float* __restrict__ Q,
                             const float* __restrict__ H,
                             const float* __restrict__ Wa,
                             const float* __restrict__ ba,
                             float* __restrict__ S) {
  const int lane = threadIdx.x & 31;
  const int wave = threadIdx.x >> 5;
  const int b    = blockIdx.x;
  const int mw   = wave >> 1;           // row quadrant
  const int nw   = wave & 1;            // col quadrant

  const int half = lane >> 4;
  const int l    = lane & 15;
  const int koff = half * 2;

  const float* q0 = Q + ((size_t)b * R_ + mw * 32 + 0  + l) * D_ + koff;
  const float* q1 = Q + ((size_t)b * R_ + mw * 32 + 16 + l) * D_ + koff;
  const float* h0 = H + ((size_t)b * R_ + nw * 32 + 0  + l) * D_ + koff;
  const float* h1 = H + ((size_t)b * R_ + nw * 32 + 16 + l) * D_ + koff;

  v8f acc1[2][2] = {};   // numerator
  v8f acc2[2][2] = {};   // squared norm
  for (int k = 0; k < D_; k += 4) {
    v2f w  = *(const v2f*)(Wa + k + koff);
    v2f qa = *(const v2f*)(q0 + k);
    v2f qb = *(const v2f*)(q1 + k);
    v2f ha = *(const v2f*)(h0 + k);
    v2f hb = *(const v2f*)(h1 + k);
    v2f a1s0 = qa * w,  a1s1 = qb * w;
    v2f a2s0 = qa * qa, a2s1 = qb * qb;
    v2f b2t0 = ha * ha, b2t1 = hb * hb;
    acc1[0][0] = wmma4(a1s0, ha, acc1[0][0]);
    acc1[0][1] = wmma4(a1s0, hb, acc1[0][1]);
    acc1[1][0] = wmma4(a1s1, ha, acc1[1][0]);
    acc1[1][1] = wmma4(a1s1, hb, acc1[1][1]);
    acc2[0][0] = wmma4(a2s0, b2t0, acc2[0][0]);
    acc2[0][1] = wmma4(a2s0, b2t1, acc2[0][1]);
    acc2[1][0] = wmma4(a2s1, b2t0, acc2[1][0]);
    acc2[1][1] = wmma4(a2s1, b2t1, acc2[1][1]);
  }

  const float bav = ba[0];
#pragma unroll
  for (int s = 0; s < 2; ++s) {
#pragma unroll
    for (int t = 0; t < 2; ++t) {
#pragma unroll
      for (int j = 0; j < 8; ++j) {
        const int r = mw * 32 + s * 16 + j + half * 8;
        const int c = nw * 32 + t * 16 + l;
        float sc = acc1[s][t][j] / fmaxf(sqrtf(acc2[s][t][j]), 1e-12f) + bav;
        S[((size_t)b * R_ + r) * R_ + c] = sc;
      }
    }
  }
}

// ---------------------------------------------------------------------------
// Kernel 3: masked softmax over k<=r, in-place on S[B*R, R].
// (softmax -> tril mask -> renorm == masked softmax; normalizer cancels)
// ---------------------------------------------------------------------------
__global__ void softmax_kernel(float* __restrict__ S) {
  const int lane = threadIdx.x & 31;
  const int wave = threadIdx.x >> 5;
  const int row  = blockIdx.x * 4 + wave;   // b*R + r
  const int r    = row & (R_ - 1);

  float* srow = S + (size_t)row * R_;
  const float s0 = srow[lane];
  const float s1 = srow[lane + 32];
  const float NEG = -3.0e38f;

  float m = fmaxf((lane <= r) ? s0 : NEG, (lane + 32 <= r) ? s1 : NEG);
#pragma unroll
  for (int off = 16; off > 0; off >>= 1)
    m = fmaxf(m, __shfl_xor(m, off, 32));

  const float e0 = (lane <= r)      ? expf(s0 - m) : 0.0f;
  const float e1 = (lane + 32 <= r) ? expf(s1 - m) : 0.0f;
  float sum = e0 + e1;
#pragma unroll
  for (int off = 16; off > 0; off >>= 1)
    sum += __shfl_xor(sum, off, 32);

  const float inv = 1.0f / sum;
  srow[lane]      = e0 * inv;
  srow[lane + 32] = e1 * inv;
}

// ---------------------------------------------------------------------------
// Kernel 4: feat[b] = att[b] (64x64) @ hist[b] (64x1024).
// Each wave computes a 32x32 output block (2x2 WMMA subtiles), K=64.
// ---------------------------------------------------------------------------
__global__ void feat_kernel(const float* __restrict__ Att,
                            const float* __restrict__ Hst,
                            float* __restrict__ Out) {
  const int lane = threadIdx.x & 31;
  const int wave = threadIdx.x >> 5;
  const int tile = blockIdx.x * 4 + wave;
  const int NT   = IN_ / 32;               // 32 col-blocks of 32
  const int b    = tile / (2 * NT);
  const int rem  = tile % (2 * NT);
  const int r0   = (rem / NT) * 32;
  const int n0   = (rem % NT) * 32;

  const int half = lane >> 4;
  const int l    = lane & 15;
  const int koff = half * 2;

  const float* a0 = Att + ((size_t)b * R_ + r0 + 0  + l) * R_ + koff;
  const float* a1 = Att + ((size_t)b * R_ + r0 + 16 + l) * R_ + koff;
  const float* hb = Hst + (size_t)b * R_ * IN_;

  v8f acc[2][2] = {};
  for (int k = 0; k < R_; k += 4) {
    v2f av0 = *(const v2f*)(a0 + k);
    v2f av1 = *(const v2f*)(a1 + k);
#pragma unroll
    for (int t = 0; t < 2; ++t) {
      v2f bv;
      bv.x = hb[(size_t)(k + koff)     * IN_ + n0 + t * 16 + l];
      bv.y = hb[(size_t)(k + koff + 1) * IN_ + n0 + t * 16 + l];
      acc[0][t] = wmma4(av0, bv, acc[0][t]);
      acc[1][t] = wmma4(av1, bv, acc[1][t]);
    }
  }

#pragma unroll
  for (int s = 0; s < 2; ++s) {
#pragma unroll
    for (int t = 0; t < 2; ++t) {
#pragma unroll
      for (int j = 0; j < 8; ++j) {
        Out[((size_t)b * R_ + r0 + s * 16 + j + half * 8) * IN_
            + n0 + t * 16 + l] = acc[s][t][j];
      }
    }
  }
}

// ---------------------------------------------------------------------------
extern "C" void kernel_launch(void* const* d_in, const int* in_sizes, int n_in,
                              void* d_out, int out_size, void* d_ws, size_t ws_size,
                              hipStream_t stream) {
  (void)in_sizes; (void)n_in; (void)out_size; (void)ws_size;

  const float* hist = (const float*)d_in[0];
  const float* ques = (const float*)d_in[1];
  const float* Wh_y = (const float*)d_in[2];
  const float* bh_y = (const float*)d_in[3];
  const float* Wh_g = (const float*)d_in[4];
  const float* bh_g = (const float*)d_in[5];
  const float* Wq_y = (const float*)d_in[6];
  const float* bq_y = (const float*)d_in[7];
  const float* Wq_g = (const float*)d_in[8];
  const float* bq_g = (const float*)d_in[9];
  const float* Wa   = (const float*)d_in[10];
  const float* ba   = (const float*)d_in[11];

  // Workspace (floats): h [8192,512] | q [8192,512] | att [128,64,64]
  float* h   = (float*)d_ws;
  float* q   = h + (size_t)B_ * R_ * D_;
  float* att = q + (size_t)B_ * R_ * D_;
  float* out = (float*)d_out;

  const dim3 blk(128);                       // 4 waves

  // gated GEMMs: grid = (n-blocks=16, m-blocks=128)
  hipLaunchKernelGGL(gated_gemm_kernel, dim3(D_ / 32, (B_ * R_) / 64), blk, 0,
                     stream, hist, Wh_y, bh_y, Wh_g, bh_g, h);
  hipLaunchKernelGGL(gated_gemm_kernel, dim3(D_ / 32, (B_ * R_) / 64), blk, 0,
                     stream, ques, Wq_y, bq_y, Wq_g, bq_g, q);

  // scores: one block per batch element
  hipLaunchKernelGGL(score_kernel, dim3(B_), blk, 0, stream, q, h, Wa, ba, att);

  // masked softmax: one wave per row
  hipLaunchKernelGGL(softmax_kernel, dim3((B_ * R_) / 4), blk, 0, stream, att);

  // feat: 2 row-blocks x 32 col-blocks per b = 64 waves/b
  hipLaunchKernelGGL(feat_kernel, dim3(B_ * 2 * (IN_ / 32) / 4), blk, 0, stream,
                     att, hist, out);
}